// HashedInterpolator_2989297238758
// MI455X (gfx1250) — compile-verified
//
#include <hip/hip_runtime.h>
#include <cstdint>

#define TPB 256
#define GRID_F 512.0f
#define H_MASK 0x7FFFFu
#define PR1 19349663u
#define PR2 83492791u

typedef float v4f __attribute__((ext_vector_type(4)));

// Faithful reimplementation of the reference math for one position.
// coef[...,0] = (p - lower/512)*512 (bit==0), coef[...,1] = ((lower+1)/512 - p)*512 (bit==1).
// hash = (x ^ y*PR1 ^ z*PR2) & 0x7FFFF with uint32 wrapping (== int32 mul + python-mod 2^19).
__device__ __forceinline__ v4f hg_interp(float p0, float p1, float p2,
                                         const v4f* __restrict__ t4) {
  const float inv = 1.0f / GRID_F;
  float l0 = floorf(p0 * GRID_F);
  float l1 = floorf(p1 * GRID_F);
  float l2 = floorf(p2 * GRID_F);
  float f0 = (p0 - l0 * inv) * GRID_F, g0 = ((l0 + 1.0f) * inv - p0) * GRID_F;
  float f1 = (p1 - l1 * inv) * GRID_F, g1 = ((l1 + 1.0f) * inv - p1) * GRID_F;
  float f2 = (p2 - l2 * inv) * GRID_F, g2 = ((l2 + 1.0f) * inv - p2) * GRID_F;

  uint32_t hx0 = (uint32_t)(int)l0,        hx1 = hx0 + 1u;
  uint32_t hy0 = (uint32_t)(int)l1 * PR1,  hy1 = hy0 + PR1;
  uint32_t hz0 = (uint32_t)(int)l2 * PR2,  hz1 = hz0 + PR2;

  uint32_t hx[2] = {hx0, hx1}, hy[2] = {hy0, hy1}, hz[2] = {hz0, hz1};
  float    w0[2] = {f0, g0},   w1[2] = {f1, g1},   w2[2] = {f2, g2};

  v4f acc = (v4f)0.0f;
#pragma unroll
  for (int c = 0; c < 8; ++c) {
    const int b0 = (c >> 2) & 1, b1 = (c >> 1) & 1, b2 = c & 1;
    uint32_t h = (hx[b0] ^ hy[b1] ^ hz[b2]) & H_MASK;
    v4f v = t4[h];                     // global_load_b128 gather, RT policy (keep table in L2)
    float w = w0[b0] * w1[b1] * w2[b2];
    acc += w * v;                      // 4x v_fmac_f32
  }
  return acc;
}

__global__ __launch_bounds__(TPB) void HashedInterpolator_2989297238758_kernel(
    const float* __restrict__ pos, const float* __restrict__ tbl,
    v4f* __restrict__ out, int n, int nTiles) {
  // Double-buffered position staging: 2 x 256 positions x 3 floats = 6144 B LDS.
  __shared__ float sp[2][TPB * 3];
  const int tid = threadIdx.x;
  const v4f* t4 = (const v4f*)tbl;

  int tile = blockIdx.x;
  int cur = 0;

  // Prologue: stage first tile (192 lanes x b128 = 3072 B), wait, publish.
  if (tile < nTiles) {
    if (tid < 192) {
      uint32_t lds = (uint32_t)(uintptr_t)(&sp[0][0]) + (uint32_t)(tid * 16);
      uint32_t go  = (uint32_t)(tile * (TPB * 12) + tid * 16);
      asm volatile("global_load_async_to_lds_b128 %0, %1, %2 th:TH_LOAD_NT"
                   :: "v"(lds), "v"(go), "s"(pos) : "memory");
    }
    asm volatile("s_wait_asynccnt 0x0" ::: "memory");
    __syncthreads();
  }

  for (; tile < nTiles; tile += gridDim.x) {
    // Kick off async copy of the next tile into the other buffer (overlaps
    // with this tile's hash/gather/FMA work). NT hint: streamed once, do not
    // displace the L2-resident hash table.
    int nxt = tile + gridDim.x;
    if (nxt < nTiles && tid < 192) {
      uint32_t lds = (uint32_t)(uintptr_t)(&sp[cur ^ 1][0]) + (uint32_t)(tid * 16);
      uint32_t go  = (uint32_t)(nxt * (TPB * 12) + tid * 16);
      asm volatile("global_load_async_to_lds_b128 %0, %1, %2 th:TH_LOAD_NT"
                   :: "v"(lds), "v"(go), "s"(pos) : "memory");
    }

    // Compute current tile from LDS (stride-3 ds_load_b96, bank-conflict free).
    float p0 = sp[cur][3 * tid + 0];
    float p1 = sp[cur][3 * tid + 1];
    float p2 = sp[cur][3 * tid + 2];
    v4f acc = hg_interp(p0, p1, p2, t4);
    // Write-once output: non-temporal b128 store, keep L2 for the table.
    __builtin_nontemporal_store(acc, &out[tile * TPB + tid]);

    // Each wave waits its own ASYNCcnt; barrier publishes all waves' staging.
    asm volatile("s_wait_asynccnt 0x0" ::: "memory");
    __syncthreads();
    cur ^= 1;
  }

  // Tail (n not a multiple of TPB) — direct path, no staging.
  for (int i = nTiles * TPB + blockIdx.x * TPB + tid; i < n; i += gridDim.x * TPB) {
    float p0 = pos[3 * i + 0];
    float p1 = pos[3 * i + 1];
    float p2 = pos[3 * i + 2];
    v4f acc = hg_interp(p0, p1, p2, t4);
    __builtin_nontemporal_store(acc, &out[i]);
  }
}

extern "C" void kernel_launch(void* const* d_in, const int* in_sizes, int n_in,
                              void* d_out, int out_size, void* d_ws, size_t ws_size,
                              hipStream_t stream) {
  const float* pos = (const float*)d_in[0];   // (BATCH, 3) f32
  const float* tbl = (const float*)d_in[1];   // (524288, 4) f32
  v4f* out = (v4f*)d_out;                     // (BATCH, 4) f32
  (void)d_ws; (void)ws_size; (void)n_in; (void)out_size;

  int n = in_sizes[0] / 3;
  if (n <= 0) return;
  int nTiles = n / TPB;                       // 8192 for BATCH=2M
  int blocks = nTiles > 0 ? (nTiles < 4096 ? nTiles : 4096) : 1;
  HashedInterpolator_2989297238758_kernel<<<blocks, TPB, 0, stream>>>(pos, tbl, out, n, nTiles);
}